// ProposalLayer_7584912245185
// MI455X (gfx1250) — compile-verified
//
#include <hip/hip_runtime.h>
#include <hip/hip_bf16.h>

#define BATCH   8
#define NANCH   131072
#define TOPK    6000
#define NP      6016          // TOPK padded to multiple of 32
#define NWORDS  188           // NP / 32
#define CAND    8192
#define NBUCK   32768
#define PROP    1000
#define NMS_T   0.7f
#define AREAS   6080          // padded area stride (avoid OOB on speculative lane loads)

typedef float v2f __attribute__((ext_vector_type(2)));
typedef float v8f __attribute__((ext_vector_type(8)));

// ---------------- init workspace (deterministic per call) ----------------
__global__ void init_ws(unsigned* hist, unsigned* cnt, float* cs, unsigned* ci) {
    int g = blockIdx.x * blockDim.x + threadIdx.x;
    if (g < BATCH * NBUCK) hist[g] = 0u;
    if (g < BATCH * CAND) { cs[g] = __uint_as_float(0xFF800000u); ci[g] = 0u; }  // -inf
    if (g < BATCH) cnt[g] = 0u;
}

// ---------------- pass 1: 32768-bucket histogram of score float bits ----------------
__global__ void hist_kernel(const float* __restrict__ probs, unsigned* __restrict__ hist) {
    int b = blockIdx.y;
    int t = blockIdx.x * blockDim.x + threadIdx.x;              // t < NANCH
    float s = probs[(((size_t)b * NANCH + t) << 1) + 1];
    unsigned bk = __float_as_uint(s) >> 17;                     // positive floats: monotonic
    atomicAdd(&hist[b * NBUCK + bk], 1u);
}

// ---------------- pass 2: find threshold bucket (suffix scan) ----------------
__global__ __launch_bounds__(1024)
void find_thresh(const unsigned* __restrict__ hist, unsigned* __restrict__ tb) {
    int b = blockIdx.x, tid = threadIdx.x;
    const unsigned* h = hist + (size_t)b * NBUCK;
    __shared__ unsigned ssum[1024];
    unsigned csum = 0;
    int base = tid * 32;
    #pragma unroll 4
    for (int q = 0; q < 32; ++q) csum += h[base + q];
    ssum[tid] = csum;
    __syncthreads();
    for (int off = 1; off < 1024; off <<= 1) {                  // inclusive suffix sum
        unsigned v = (tid + off < 1024) ? ssum[tid + off] : 0u;
        __syncthreads();
        ssum[tid] += v;
        __syncthreads();
    }
    unsigned suff_incl = ssum[tid];
    unsigned suff_above = suff_incl - csum;
    if (suff_above < TOPK && suff_incl >= TOPK) {               // exactly one thread
        unsigned run = suff_above;
        int bk = base + 31;
        for (; bk >= base; --bk) {
            run += h[bk];
            if (run >= TOPK) break;
        }
        tb[b] = (unsigned)bk;
    }
}

// ---------------- pass 3: compact candidates >= threshold bucket ----------------
__global__ void compact_kernel(const float* __restrict__ probs, const unsigned* __restrict__ tb,
                               unsigned* __restrict__ cnt, float* __restrict__ cs,
                               unsigned* __restrict__ ci) {
    int b = blockIdx.y;
    int t = blockIdx.x * blockDim.x + threadIdx.x;
    float s = probs[(((size_t)b * NANCH + t) << 1) + 1];
    unsigned bk = __float_as_uint(s) >> 17;
    if (bk >= tb[b]) {
        unsigned slot = atomicAdd(&cnt[b], 1u);
        if (slot < CAND) {
            cs[(size_t)b * CAND + slot] = s;
            ci[(size_t)b * CAND + slot] = (unsigned)t;
        }
    }
}

// ---------------- pass 4: exact bitonic sort of 8192 candidates in LDS ----------------
// key = (monotone(score) << 32) | ~idx  -> sort descending => score desc, idx asc ties
__global__ __launch_bounds__(1024)
void sort_cand(float* __restrict__ cs, unsigned* __restrict__ ci) {
    int b = blockIdx.x, tid = threadIdx.x;
    __shared__ unsigned long long keys[CAND];                    // 64 KB
    float* csb = cs + (size_t)b * CAND;
    unsigned* cib = ci + (size_t)b * CAND;
    for (int e = tid; e < CAND; e += 1024) {
        unsigned fb = __float_as_uint(csb[e]);
        unsigned m = (fb & 0x80000000u) ? 0xFFFFFFFFu : 0x80000000u;
        keys[e] = ((unsigned long long)(fb ^ m) << 32) | (unsigned)(~cib[e]);
    }
    __syncthreads();
    for (int k = 2; k <= CAND; k <<= 1) {
        for (int j = k >> 1; j > 0; j >>= 1) {
            for (int e = tid; e < CAND; e += 1024) {
                int p = e ^ j;
                if (p > e) {
                    bool up = ((e & k) == 0);
                    unsigned long long a = keys[e], bb = keys[p];
                    bool sw = up ? (a < bb) : (a > bb);           // descending overall
                    if (sw) { keys[e] = bb; keys[p] = a; }
                }
            }
            __syncthreads();
        }
    }
    for (int e = tid; e < NP; e += 1024) {
        unsigned long long kk = keys[e];
        unsigned hi = (unsigned)(kk >> 32);
        unsigned m = (hi & 0x80000000u) ? 0x80000000u : 0xFFFFFFFFu;
        csb[e] = __uint_as_float(hi ^ m);
        cib[e] = ~(unsigned)kk;
    }
}

// ---------------- pass 5: apply deltas, clip, precompute areas ----------------
__global__ void box_kernel(const float* __restrict__ bbox, const float* __restrict__ anchors,
                           const unsigned* __restrict__ ci, float* __restrict__ boxes,
                           float* __restrict__ area) {
    int g = blockIdx.x * blockDim.x + threadIdx.x;
    if (g >= BATCH * NP) return;
    int b = g / NP, i = g % NP;
    unsigned idx = ci[(size_t)b * CAND + i];
    if (idx >= NANCH) idx = 0;
    size_t o = (((size_t)b * NANCH + idx) << 2);
    float d0 = bbox[o + 0] * 0.1f, d1 = bbox[o + 1] * 0.1f;
    float d2 = bbox[o + 2] * 0.2f, d3 = bbox[o + 3] * 0.2f;
    float a0 = anchors[o + 0], a1 = anchors[o + 1], a2 = anchors[o + 2], a3 = anchors[o + 3];
    float h = a2 - a0, w = a3 - a1;
    float cy = a0 + 0.5f * h + d0 * h;
    float cx = a1 + 0.5f * w + d1 * w;
    h *= __expf(d2);
    w *= __expf(d3);
    float y1 = cy - 0.5f * h, x1 = cx - 0.5f * w;
    float y2 = y1 + h, x2 = x1 + w;
    y1 = fminf(fmaxf(y1, 0.f), 1.f);
    x1 = fminf(fmaxf(x1, 0.f), 1.f);
    y2 = fminf(fmaxf(y2, 0.f), 1.f);
    x2 = fminf(fmaxf(x2, 0.f), 1.f);
    size_t ob = ((size_t)b * NP + i) << 2;
    boxes[ob + 0] = y1; boxes[ob + 1] = x1; boxes[ob + 2] = y2; boxes[ob + 3] = x2;
    area[(size_t)b * AREAS + i] = (y2 - y1) * (x2 - x1);
}

// ---------------- pass 6: lower-triangle IoU bit-mask, WMMA for union term ----------------
// One wave32 per 16-row x 32-col tile. V_WMMA_F32_16X16X4_F32 computes the rank-2
// outer product area_i + area_j (the separable part of the union) on the matrix pipe,
// co-executing with the VALU min/max intersection math. Ballot predicates lower into
// v_cmp SGPR masks; each lane<16 latches its row's word and one strided store emits
// the whole 16-row x 32-col bit tile.
__global__ __launch_bounds__(32)
void iou_mask(const float* __restrict__ boxes, const float* __restrict__ area,
              unsigned* __restrict__ mask) {
    int tx = blockIdx.x;              // column word index (32 cols)
    int ty = blockIdx.y;              // row tile (16 rows)
    int b  = blockIdx.z;
    if (32 * tx > 16 * ty + 15) return;               // triangle: only j <= i_max needed
    int lane = threadIdx.x;
    int i0 = ty * 16, j0 = tx * 32;

    const float4* boxB = ((const float4*)boxes) + (size_t)b * NP;
    const float*  arB  = area + (size_t)b * AREAS;

    __shared__ float4 bi[16];
    if (lane < 16) bi[lane] = boxB[i0 + lane];
    __syncthreads();

    bool lo = (lane < 16);
    // A: 16x4, lanes 0-15 hold K=0..1 -> [area_i, 1]; lanes 16-31 hold K=2..3 -> zeros
    v2f amat; amat.x = lo ? arB[i0 + lane] : 0.f;      amat.y = lo ? 1.f : 0.f;
    // B: 4x16, lanes 0-15 hold K=0..1 -> [1, area_j]; lanes 16-31 -> zeros
    v2f b0m;  b0m.x = lo ? 1.f : 0.f;                  b0m.y = lo ? arB[j0 + lane] : 0.f;
    v2f b1m;  b1m.x = b0m.x;                           b1m.y = lo ? arB[j0 + 16 + lane] : 0.f;

    v8f zc = {0.f, 0.f, 0.f, 0.f, 0.f, 0.f, 0.f, 0.f};
#if __has_builtin(__builtin_amdgcn_wmma_f32_16x16x4_f32)
    v8f c0 = __builtin_amdgcn_wmma_f32_16x16x4_f32(false, amat, false, b0m, (short)0, zc, false, false);
    v8f c1 = __builtin_amdgcn_wmma_f32_16x16x4_f32(false, amat, false, b1m, (short)0, zc, false, false);
#else
    v8f c0 = zc, c1 = zc;   // fallback (should not trigger on gfx1250 toolchains)
    #pragma unroll
    for (int r = 0; r < 8; ++r) {
        c0[r] = arB[i0 + r + ((lane >> 4) << 3)] + arB[j0 + (lane & 15)];
        c1[r] = arB[i0 + r + ((lane >> 4) << 3)] + arB[j0 + 16 + (lane & 15)];
    }
#endif
    // D layout: lanes 0-15 -> N=lane, M=r ; lanes 16-31 -> N=lane-16, M=r+8
    int n = lane & 15;
    float4 bj0 = boxB[j0 + n];
    float4 bj1 = boxB[j0 + 16 + n];
    int mOff = (lane >> 4) << 3;

    unsigned myw = 0u;                // lane r: w_lo(row i0+r); lane r+8: w_hi(row i0+r+8)
    #pragma unroll
    for (int r = 0; r < 8; ++r) {
        float4 bx = bi[r + mOff];
        float iy0 = fminf(bx.z, bj0.z) - fmaxf(bx.x, bj0.x);
        float ix0 = fminf(bx.w, bj0.w) - fmaxf(bx.y, bj0.y);
        float in0 = fmaxf(iy0, 0.f) * fmaxf(ix0, 0.f);
        bool p0 = in0 > NMS_T * (c0[r] - in0);          // iou > T  <=>  inter > T*union
        float iy1 = fminf(bx.z, bj1.z) - fmaxf(bx.x, bj1.x);
        float ix1 = fminf(bx.w, bj1.w) - fmaxf(bx.y, bj1.y);
        float in1 = fmaxf(iy1, 0.f) * fmaxf(ix1, 0.f);
        bool p1 = in1 > NMS_T * (c1[r] - in1);
        unsigned bal0 = (unsigned)__ballot(p0);         // wave-uniform SGPR mask
        unsigned bal1 = (unsigned)__ballot(p1);
        unsigned wlo = (bal0 & 0xFFFFu) | ((bal1 & 0xFFFFu) << 16);
        unsigned whi = ((bal0 >> 16) & 0xFFFFu) | (bal1 & 0xFFFF0000u);
        myw = (lane == r)       ? wlo : myw;            // v_cndmask from SGPR
        myw = (lane == (r + 8)) ? whi : myw;
    }
    unsigned* mrow = mask + ((size_t)b * TOPK + i0) * NWORDS + tx;
    if (lane < 16) mrow[(size_t)lane * NWORDS] = myw;   // one strided 16-lane store
}

// ---------------- pass 7: sequential greedy scan (one wave per batch) ----------------
// keep[i] = !any_{j<i kept} iou(i,j) > T ; masks are L2-resident (36 MB << 192 MB L2).
__global__ __launch_bounds__(32)
void nms_scan(const unsigned* __restrict__ mask, const float4* __restrict__ boxes4,
              float4* __restrict__ out) {
    int b = blockIdx.x, lane = threadIdx.x;
    __shared__ unsigned keepw[NWORDS];
    for (int w = lane; w < NWORDS; w += 32) keepw[w] = 0u;
    __syncthreads();
    const unsigned* mb = mask + (size_t)b * TOPK * NWORDS;
    int count = 0;
    for (int i = 0; i < TOPK; ++i) {
        const unsigned* row = mb + (size_t)i * NWORDS;
        if (i + 1 < TOPK) __builtin_prefetch(row + NWORDS, 0, 0);   // global_prefetch_b8
        int lastw = i >> 5, lastb = i & 31;
        bool sup = false;
        for (int w = lane; w <= lastw; w += 32) {
            unsigned m = row[w] & keepw[w];
            if (w == lastw) m &= (lastb ? ((1u << lastb) - 1u) : 0u);
            sup |= (m != 0u);
        }
        sup = __any(sup);
        if (!sup) {
            if (lane == 0) keepw[i >> 5] |= (1u << (i & 31));
            __syncthreads();
            if (lane == 0 && count < PROP) out[(size_t)b * PROP + count] = boxes4[(size_t)b * NP + i];
            ++count;
        }
    }
    int start = (count < PROP) ? count : PROP;
    float4 z4; z4.x = 0.f; z4.y = 0.f; z4.z = 0.f; z4.w = 0.f;
    for (int k = start + lane; k < PROP; k += 32) out[(size_t)b * PROP + k] = z4;
}

// ---------------- host-side launch ----------------
extern "C" void kernel_launch(void* const* d_in, const int* in_sizes, int n_in,
                              void* d_out, int out_size, void* d_ws, size_t ws_size,
                              hipStream_t stream) {
    (void)in_sizes; (void)n_in; (void)out_size; (void)ws_size;
    const float* probs   = (const float*)d_in[0];   // [8,131072,2]
    const float* bbox    = (const float*)d_in[1];   // [8,131072,4]
    const float* anchors = (const float*)d_in[2];   // [8,131072,4]
    float4* out4 = (float4*)d_out;                  // [8,1000,4]

    char* w = (char*)d_ws;
    unsigned* hist  = (unsigned*)w;  w += (size_t)BATCH * NBUCK * 4;   // 1 MB
    unsigned* tb    = (unsigned*)w;  w += 256;
    unsigned* cnt   = (unsigned*)w;  w += 256;
    float*    cs    = (float*)w;     w += (size_t)BATCH * CAND * 4;    // 256 KB
    unsigned* ci    = (unsigned*)w;  w += (size_t)BATCH * CAND * 4;    // 256 KB
    float*    boxes = (float*)w;     w += (size_t)BATCH * NP * 16;     // 752 KB
    float*    area  = (float*)w;     w += (size_t)BATCH * AREAS * 4;   // 190 KB
    unsigned* mask  = (unsigned*)w;                                    // 36 MB

    init_ws<<<(BATCH * NBUCK + 255) / 256, 256, 0, stream>>>(hist, cnt, cs, ci);
    hist_kernel<<<dim3(NANCH / 256, BATCH), 256, 0, stream>>>(probs, hist);
    find_thresh<<<BATCH, 1024, 0, stream>>>(hist, tb);
    compact_kernel<<<dim3(NANCH / 256, BATCH), 256, 0, stream>>>(probs, tb, cnt, cs, ci);
    sort_cand<<<BATCH, 1024, 0, stream>>>(cs, ci);
    box_kernel<<<(BATCH * NP + 255) / 256, 256, 0, stream>>>(bbox, anchors, ci, boxes, area);
    iou_mask<<<dim3(NWORDS, TOPK / 16, BATCH), 32, 0, stream>>>(boxes, area, mask);
    nms_scan<<<BATCH, 32, 0, stream>>>(mask, (const float4*)boxes, out4);
}